// MolGANDiscriminator_523986010265
// MI455X (gfx1250) — compile-verified
//
#include <hip/hip_runtime.h>
#include <hip/hip_bf16.h>

typedef __attribute__((ext_vector_type(2))) float v2f;
typedef __attribute__((ext_vector_type(8))) float v8f;

#define B_MOL   32768
#define NATOM   9
#define NREL    4
#define LRELU   0.2f
#define BNEPS   1e-5f
#define MOLS_WG 16
#define ROWS    144          // MOLS_WG * NATOM
#define NWG     2048         // B_MOL / MOLS_WG

// ---------------------------------------------------------------------------
// Per-layer RGCN kernel.
//   h' = H @ Ws + sum_r (adj_r @ H) @ Wr_r + bias     (per molecule)
// computed as nine 16-row WMMA tiles over 16 molecules per workgroup.
// BN(prev stats)+LeakyReLU of the previous layer is fused into the input load.
// Per-atom sum / sumsq (post-bias, pre-BN) are written as per-WG partials.
// ---------------------------------------------------------------------------
template <int DIN, int DOUT, bool FIRST>
__global__ __launch_bounds__(32 * ((DOUT / 16) < 8 ? (DOUT / 16) : 8))
void rgcn_layer_kernel(const float* __restrict__ xin,       // [B,9,DIN] (raw prev h or atom_feats)
                       const float* __restrict__ adj,       // [B,4,9,9]
                       const float* __restrict__ wself,     // [DIN,DOUT]
                       const float* __restrict__ wrel,      // [4,DIN,DOUT]
                       const float* __restrict__ bias,      // [DOUT]
                       const float* __restrict__ gprev,     // [9]  (prev-layer gamma)
                       const float* __restrict__ beprev,    // [9]  (prev-layer beta)
                       const float* __restrict__ statsprev, // [18] sum[9],sumsq[9]
                       float prevCnt,
                       float* __restrict__ hout,            // [B,9,DOUT] raw (pre-BN)
                       float* __restrict__ partials)        // [NWG,18]
{
    constexpr int KPAD = (DIN + 3) & ~3;                 // pad K to multiple of 4
    constexpr int KCH  = (KPAD < 32) ? KPAD : 32;        // K chunk held in LDS
    constexpr int HSC  = KCH + 1;                        // padded LDS row stride
    constexpr int NCT  = DOUT / 16;                      // 16-wide column tiles
    constexpr int NTH  = 32 * (NCT < 8 ? NCT : 8);       // threads (all waves active)
    constexpr int CPW  = NCT / (NTH / 32);               // col tiles per wave
    constexpr bool KALIGNED = (KPAD == DIN);             // no K-padding guards needed

    __shared__ float Hs [ROWS * HSC];
    __shared__ float AHs[ROWS * HSC];
    __shared__ float Adjs[MOLS_WG * NREL * 81];
    __shared__ float lsum[NATOM], lsq[NATOM];
    __shared__ float bnsc[NATOM], bnsh[NATOM];

    const int tid   = threadIdx.x;
    const int lane  = tid & 31;
    const int wave  = tid >> 5;
    const int wg    = blockIdx.x;
    const int m16   = lane & 15;
    const int khalf = (lane >> 4) << 1;   // 0 or 2 (f32 WMMA A/B lane layout)

    if (tid < NATOM) { lsum[tid] = 0.f; lsq[tid] = 0.f; }
    if constexpr (!FIRST) {
        if (tid < NATOM) {
            float mu  = statsprev[tid] / prevCnt;
            float var = statsprev[NATOM + tid] / prevCnt - mu * mu;
            float sc  = gprev[tid] * rsqrtf(var + BNEPS);
            bnsc[tid] = sc;
            bnsh[tid] = beprev[tid] - mu * sc;
        }
    }
    for (int idx = tid; idx < MOLS_WG * NREL * 81; idx += NTH)
        Adjs[idx] = adj[(size_t)wg * (MOLS_WG * NREL * 81) + idx];

    v8f acc[9 * CPW];
    const v8f vzero = {0.f, 0.f, 0.f, 0.f, 0.f, 0.f, 0.f, 0.f};
#pragma unroll
    for (int i = 0; i < 9 * CPW; ++i) acc[i] = vzero;

    // GEMM over one K-chunk of an LDS-resident A against global weights W.
    auto gemm_chunk = [&](const float* Alds, const float* __restrict__ W, int kb) {
        for (int ks = 0; ks < KCH; ks += 4) {
            v2f af[9];
#pragma unroll
            for (int rt = 0; rt < 9; ++rt) {
                const float* ap = &Alds[(rt * 16 + m16) * HSC + ks + khalf];
                af[rt] = v2f{ap[0], ap[1]};
            }
#pragma unroll
            for (int c = 0; c < CPW; ++c) {
                const int ct  = wave * CPW + c;          // always < NCT by construction
                const int col = ct * 16 + m16;
                const int kr  = kb + ks + khalf;
                float b0, b1;
                if constexpr (KALIGNED) {
                    // kr+1 <= kb + KCH - 1 < DIN always: unconditional loads,
                    // no exec-masked branches in the hot path.
                    b0 = W[(size_t)kr * DOUT + col];
                    b1 = W[(size_t)(kr + 1) * DOUT + col];
                } else {
                    b0 = (kr     < DIN) ? W[(size_t)kr * DOUT + col]       : 0.f;
                    b1 = (kr + 1 < DIN) ? W[(size_t)(kr + 1) * DOUT + col] : 0.f;
                }
                v2f bf = v2f{b0, b1};
#pragma unroll
                for (int rt = 0; rt < 9; ++rt)
                    acc[rt * CPW + c] = __builtin_amdgcn_wmma_f32_16x16x4_f32(
                        false, af[rt], false, bf, (short)0, acc[rt * CPW + c],
                        false, false);
            }
        }
    };

    for (int kb = 0; kb < KPAD; kb += KCH) {
        __syncthreads();   // prior chunk fully consumed; bnsc/Adjs visible
        // Load H chunk, fusing BN(prev)+LeakyReLU; zero-pad k >= DIN.
        for (int idx = tid; idx < ROWS * KCH; idx += NTH) {
            int row = idx / KCH;
            int kk  = idx - row * KCH;
            int k   = kb + kk;
            float v;
            if (KALIGNED || k < DIN) {
                v = xin[(size_t)wg * ROWS * DIN + (size_t)row * DIN + k];
                if constexpr (!FIRST) {
                    int atom = row % NATOM;
                    v = v * bnsc[atom] + bnsh[atom];
                    v = v > 0.f ? v : LRELU * v;
                }
            } else {
                v = 0.f;
            }
            Hs[row * HSC + kk] = v;
        }
        __syncthreads();
        gemm_chunk(Hs, wself, kb);                       // self-transform segment
        for (int r = 0; r < NREL; ++r) {
            __syncthreads();                             // AHs free for reuse
            for (int idx = tid; idx < ROWS * KCH; idx += NTH) {
                int row  = idx / KCH;
                int kk   = idx - row * KCH;
                int mol  = row / NATOM;
                int atom = row - mol * NATOM;
                const float* arow = &Adjs[(mol * NREL + r) * 81 + atom * 9];
                const float* hb   = &Hs[(mol * NATOM) * HSC + kk];
                float s = 0.f;
#pragma unroll
                for (int n = 0; n < NATOM; ++n) s += arow[n] * hb[n * HSC];
                AHs[row * HSC + kk] = s;
            }
            __syncthreads();
            gemm_chunk(AHs, wrel + (size_t)r * DIN * DOUT, kb);
        }
    }

    // Epilogue: bias, store raw h', accumulate per-atom stats.
#pragma unroll
    for (int rt = 0; rt < 9; ++rt) {
#pragma unroll
        for (int c = 0; c < CPW; ++c) {
            const int ct  = wave * CPW + c;
            const int col = ct * 16 + m16;
            const float bc = bias[col];
#pragma unroll
            for (int v = 0; v < 8; ++v) {
                int lr    = rt * 16 + v + ((lane >> 4) << 3);
                float val = acc[rt * CPW + c][v] + bc;
                hout[(size_t)(wg * ROWS + lr) * DOUT + col] = val;
                int atom = lr % NATOM;
                atomicAdd(&lsum[atom], val);
                atomicAdd(&lsq[atom], val * val);
            }
        }
    }
    __syncthreads();
    if (tid < NATOM) {
        partials[(size_t)wg * 18 + tid]         = lsum[tid];
        partials[(size_t)wg * 18 + NATOM + tid] = lsq[tid];
    }
}

// Deterministic tree reduction of per-WG stat partials -> 18 floats.
__global__ __launch_bounds__(256)
void reduce_stats_kernel(const float* __restrict__ part, float* __restrict__ stats)
{
    __shared__ float red[256];
    const int tid = threadIdx.x;
    for (int j = 0; j < 18; ++j) {
        float s = 0.f;
        for (int i = tid; i < NWG; i += 256) s += part[(size_t)i * 18 + j];
        red[tid] = s;
        __syncthreads();
        for (int off = 128; off > 0; off >>= 1) {
            if (tid < off) red[tid] += red[tid + off];
            __syncthreads();
        }
        if (tid == 0) stats[j] = red[0];
        __syncthreads();
    }
}

// Final: BN2 + LeakyReLU + masked atom-sum + FC(256->1). One wave per molecule.
__global__ __launch_bounds__(256)
void final_kernel(const float* __restrict__ h2, const float* __restrict__ mask,
                  const float* __restrict__ g2, const float* __restrict__ be2,
                  const float* __restrict__ stats2,
                  const float* __restrict__ fcw, const float* __restrict__ fcb,
                  float* __restrict__ out)
{
    const int mol  = blockIdx.x * 8 + (threadIdx.x >> 5);
    const int lane = threadIdx.x & 31;
    const float cnt = (float)B_MOL * 256.f;
    float sc[NATOM], sh[NATOM], msk[NATOM];
#pragma unroll
    for (int n = 0; n < NATOM; ++n) {
        float mu  = stats2[n] / cnt;
        float var = stats2[NATOM + n] / cnt - mu * mu;
        float s   = g2[n] * rsqrtf(var + BNEPS);
        sc[n] = s;
        sh[n] = be2[n] - mu * s;
        msk[n] = mask[(size_t)mol * NATOM + n];
    }
    float partial = 0.f;
#pragma unroll
    for (int eb = 0; eb < 8; ++eb) {
        int e = eb * 32 + lane;
        float gs = 0.f;
#pragma unroll
        for (int n = 0; n < NATOM; ++n) {
            float v = h2[(size_t)(mol * NATOM + n) * 256 + e];
            v = v * sc[n] + sh[n];
            v = v > 0.f ? v : LRELU * v;
            gs += v * msk[n];
        }
        partial += gs * fcw[e];
    }
#pragma unroll
    for (int off = 16; off > 0; off >>= 1)
        partial += __shfl_down(partial, off, 32);
    if (lane == 0) out[mol] = partial + fcb[0];
}

extern "C" void kernel_launch(void* const* d_in, const int* in_sizes, int n_in,
                              void* d_out, int out_size, void* d_ws, size_t ws_size,
                              hipStream_t stream)
{
    (void)in_sizes; (void)n_in; (void)out_size; (void)ws_size;
    const float* atom_feats = (const float*)d_in[0];
    const float* adj        = (const float*)d_in[1];
    const float* mask       = (const float*)d_in[2];
    const float* wself0 = (const float*)d_in[3];
    const float* wrel0  = (const float*)d_in[4];
    const float* b0     = (const float*)d_in[5];
    const float* g0     = (const float*)d_in[6];
    const float* be0    = (const float*)d_in[7];
    const float* wself1 = (const float*)d_in[8];
    const float* wrel1  = (const float*)d_in[9];
    const float* b1     = (const float*)d_in[10];
    const float* g1     = (const float*)d_in[11];
    const float* be1    = (const float*)d_in[12];
    const float* wself2 = (const float*)d_in[13];
    const float* wrel2  = (const float*)d_in[14];
    const float* b2     = (const float*)d_in[15];
    const float* g2     = (const float*)d_in[16];
    const float* be2    = (const float*)d_in[17];
    const float* fcw    = (const float*)d_in[18];
    const float* fcb    = (const float*)d_in[19];
    float* out = (float*)d_out;

    // Workspace layout (floats):
    float* w        = (float*)d_ws;
    float* stats0   = w + 0;      // 18 used
    float* stats1   = w + 32;
    float* stats2   = w + 64;
    float* partials = w + 128;                       // NWG*18 = 36864
    float* h0       = w + 40960;                     // [B,9,64]
    float* h1       = h0 + (size_t)B_MOL * 9 * 64;   // [B,9,128]
    float* h2       = h1 + (size_t)B_MOL * 9 * 128;  // [B,9,256]

    rgcn_layer_kernel<5, 64, true><<<NWG, 128, 0, stream>>>(
        atom_feats, adj, wself0, wrel0, b0, nullptr, nullptr, nullptr, 1.f,
        h0, partials);
    reduce_stats_kernel<<<1, 256, 0, stream>>>(partials, stats0);

    rgcn_layer_kernel<64, 128, false><<<NWG, 256, 0, stream>>>(
        h0, adj, wself1, wrel1, b1, g0, be0, stats0, (float)B_MOL * 64.f,
        h1, partials);
    reduce_stats_kernel<<<1, 256, 0, stream>>>(partials, stats1);

    rgcn_layer_kernel<128, 256, false><<<NWG, 256, 0, stream>>>(
        h1, adj, wself2, wrel2, b2, g1, be1, stats1, (float)B_MOL * 128.f,
        h2, partials);
    reduce_stats_kernel<<<1, 256, 0, stream>>>(partials, stats2);

    final_kernel<<<B_MOL / 8, 256, 0, stream>>>(
        h2, mask, g2, be2, stats2, fcw, fcb, out);
}